// MeshGNN_54408645706323
// MI455X (gfx1250) — compile-verified
//
#include <hip/hip_runtime.h>

// GraphSAGE-style GNN, 15 layers, N=200000 nodes (dim 3), E=12.8M edges.
// Memory/atomic bound: whole working set (~105MB) is resident in MI455X 192MB L2.
// Dense transform via V_WMMA_F32_16X16X4_F32 (two chained K=4 WMMAs, bias in C).

typedef __attribute__((ext_vector_type(2))) float v2f;
typedef __attribute__((ext_vector_type(8))) float v8f;

#define LAYERS 15

// ---- init: pack x [N,3] -> padded [P,4], zero agg and deg -------------------
__global__ void gnn_init(const float* __restrict__ x, float* __restrict__ xa,
                         float* __restrict__ agg, float* __restrict__ deginv,
                         int n, int p) {
  int i = blockIdx.x * blockDim.x + threadIdx.x;
  if (i >= p) return;
  float4 v = make_float4(0.f, 0.f, 0.f, 0.f);
  if (i < n) { v.x = x[3*i]; v.y = x[3*i+1]; v.z = x[3*i+2]; }
  *(float4*)(xa + 4*(size_t)i) = v;
  *(float4*)(agg + 4*(size_t)i) = make_float4(0.f, 0.f, 0.f, 0.f);
  deginv[i] = 0.0f;
}

// ---- scatter: agg[dst] += x[src]; optionally count degree (layer 0 only) ----
template<bool COUNT_DEG>
__device__ __forceinline__ void edge_op(const float* __restrict__ x,
                                        float* __restrict__ agg,
                                        float* __restrict__ deginv,
                                        int s, int d) {
  float4 xs = *(const float4*)(x + 4*(size_t)s);  // 16B gather (L2-resident)
  float* a = agg + 4*(size_t)d;
  unsafeAtomicAdd(a + 0, xs.x);                   // hw global_atomic_add_f32
  unsafeAtomicAdd(a + 1, xs.y);
  unsafeAtomicAdd(a + 2, xs.z);
  if (COUNT_DEG) unsafeAtomicAdd(deginv + d, 1.0f);
}

// 4 edges per thread: b128 streaming loads of the index arrays, 4 independent
// gathers + 12 atomics in flight per thread for memory-level parallelism.
template<bool COUNT_DEG>
__global__ void gnn_scatter(const float* __restrict__ x, const int* __restrict__ src,
                            const int* __restrict__ dst, float* __restrict__ agg,
                            float* __restrict__ deginv, int e_count) {
  int t = blockIdx.x * blockDim.x + threadIdx.x;
  int base = t * 4;
  if (base >= e_count) return;
  if (base + 3 < e_count) {
    int4 s4 = *(const int4*)(src + base);
    int4 d4 = *(const int4*)(dst + base);
    edge_op<COUNT_DEG>(x, agg, deginv, s4.x, d4.x);
    edge_op<COUNT_DEG>(x, agg, deginv, s4.y, d4.y);
    edge_op<COUNT_DEG>(x, agg, deginv, s4.z, d4.z);
    edge_op<COUNT_DEG>(x, agg, deginv, s4.w, d4.w);
  } else {
    for (int e = base; e < e_count; ++e)
      edge_op<COUNT_DEG>(x, agg, deginv, src[e], dst[e]);
  }
}

// ---- deg -> 1/max(deg,1), once ----------------------------------------------
__global__ void gnn_invdeg(float* __restrict__ deginv, int p) {
  int i = blockIdx.x * blockDim.x + threadIdx.x;
  if (i >= p) return;
  deginv[i] = 1.0f / fmaxf(deginv[i], 1.0f);
}

// ---- node update: relu(concat([x, agg/deg]) @ W + b) via WMMA f32 16x16x4 ---
// One wave handles a 16-node tile. A is 16x6 (x | mean) split into two K=4
// WMMAs; B is W (6x3) in a 4x16 tile (cols >=3 zero); C preloaded with bias.
template<bool FINAL>
__global__ void gnn_node(const float* __restrict__ xin, float* __restrict__ agg,
                         const float* __restrict__ deginv, const float* __restrict__ W,
                         const float* __restrict__ bias, float* __restrict__ xout,
                         int n_tiles, int n_nodes) {
  int gtid = blockIdx.x * blockDim.x + threadIdx.x;
  int wave = gtid >> 5;
  int lane = threadIdx.x & 31;
  if (wave >= n_tiles) return;           // uniform per wave -> EXEC all-ones below

  int m  = lane & 15;
  int hi = lane >> 4;                    // lane half selects K (A/B) / M+8 (C/D)
  int node = wave * 16 + m;

  float4 xi = *(const float4*)(xin + 4*(size_t)node);
  float4 ag = *(const float4*)(agg + 4*(size_t)node);
  float ia = deginv[node];
  float m0 = ag.x * ia, m1 = ag.y * ia, m2 = ag.z * ia;

  // A (16x4 f32 layout): VGPR0 = K0(lo)/K2(hi), VGPR1 = K1(lo)/K3(hi)
  // first WMMA: K=0..3 = {x0, x1, x2, mean0}; second: K=4,5 = {mean1, mean2}
  v2f a0, a1;
  a0.x = hi ? xi.z : xi.x;
  a0.y = hi ? m0   : xi.y;
  a1.x = hi ? 0.f  : m1;
  a1.y = hi ? 0.f  : m2;

  // B (4x16 f32 layout): VGPR0 = K0(lo)/K2(hi), VGPR1 = K1(lo)/K3(hi); N = lane&15
  float w0 = 0.f, w1 = 0.f, w2 = 0.f, w3 = 0.f, w4 = 0.f, w5 = 0.f, bn = 0.f;
  if (m < 3) {
    w0 = W[0*3+m]; w1 = W[1*3+m]; w2 = W[2*3+m];
    w3 = W[3*3+m]; w4 = W[4*3+m]; w5 = W[5*3+m];
    bn = bias[m];
  }
  v2f b0, b1;
  b0.x = hi ? w2  : w0;
  b0.y = hi ? w3  : w1;
  b1.x = hi ? 0.f : w4;
  b1.y = hi ? 0.f : w5;

  v8f c;
#pragma unroll
  for (int r = 0; r < 8; ++r) c[r] = bn;   // bias broadcast into C

  c = __builtin_amdgcn_wmma_f32_16x16x4_f32(false, a0, false, b0, (short)0, c, false, false);
  c = __builtin_amdgcn_wmma_f32_16x16x4_f32(false, a1, false, b1, (short)0, c, false, false);

  // reset agg for the next layer's scatter (one lane-half per node)
  if (!hi) *(float4*)(agg + 4*(size_t)node) = make_float4(0.f, 0.f, 0.f, 0.f);

  // D layout: lane owns column n=m, VGPR r holds row M = r + 8*hi.
  // full_tile is wave-uniform: hot path is 8 straight stores, no per-row cmpx.
  bool full_tile = (wave * 16 + 16 <= n_nodes);
  if (m < 3) {
    if (!FINAL) {
#pragma unroll
      for (int r = 0; r < 8; ++r) {
        int out_node = wave * 16 + r + 8 * hi;
        xout[(size_t)out_node * 4 + m] = fmaxf(c[r], 0.0f);
      }
    } else if (full_tile) {
#pragma unroll
      for (int r = 0; r < 8; ++r) {
        int out_node = wave * 16 + r + 8 * hi;
        xout[(size_t)out_node * 3 + m] = fmaxf(c[r], 0.0f);
      }
    } else {
#pragma unroll
      for (int r = 0; r < 8; ++r) {
        int out_node = wave * 16 + r + 8 * hi;
        if (out_node < n_nodes) xout[(size_t)out_node * 3 + m] = fmaxf(c[r], 0.0f);
      }
    }
  }
}

extern "C" void kernel_launch(void* const* d_in, const int* in_sizes, int n_in,
                              void* d_out, int out_size, void* d_ws, size_t ws_size,
                              hipStream_t stream) {
  const float* x    = (const float*)d_in[0];
  const int*   ei   = (const int*)d_in[1];
  const float* W    = (const float*)d_in[2];
  const float* bias = (const float*)d_in[3];

  int n       = in_sizes[0] / 3;
  int e_count = in_sizes[1] / 2;
  const int* src = ei;
  const int* dst = ei + (size_t)e_count;

  int n_tiles = (n + 15) / 16;
  size_t p = (size_t)n_tiles * 16;       // node count padded to 16-node tiles

  float* ws     = (float*)d_ws;
  float* xa     = ws;                    // [P,4] ping
  float* xb     = ws + 4 * p;            // [P,4] pong
  float* agg    = ws + 8 * p;            // [P,4] accumulator
  float* deginv = ws + 12 * p;           // [P]   1/max(deg,1)

  int nb = (int)((p + 255) / 256);
  int e4 = (e_count + 3) / 4;            // 4 edges per thread
  int eb = (e4 + 255) / 256;
  int tb = (n_tiles * 32 + 255) / 256;

  gnn_init<<<nb, 256, 0, stream>>>(x, xa, agg, deginv, n, (int)p);

  float* cur = xa;
  float* nxt = xb;
  for (int layer = 0; layer < LAYERS; ++layer) {
    if (layer == 0) {
      gnn_scatter<true><<<eb, 256, 0, stream>>>(cur, src, dst, agg, deginv, e_count);
      gnn_invdeg<<<nb, 256, 0, stream>>>(deginv, (int)p);
    } else {
      gnn_scatter<false><<<eb, 256, 0, stream>>>(cur, src, dst, agg, deginv, e_count);
    }
    if (layer == LAYERS - 1) {
      gnn_node<true><<<tb, 256, 0, stream>>>(cur, agg, deginv, W, bias,
                                             (float*)d_out, n_tiles, n);
    } else {
      gnn_node<false><<<tb, 256, 0, stream>>>(cur, agg, deginv, W, bias,
                                              nxt, n_tiles, n);
      float* t = cur; cur = nxt; nxt = t;
    }
  }
}